// GLM_54116587929809
// MI455X (gfx1250) — compile-verified
//
#include <hip/hip_runtime.h>
#include <hip/hip_bf16.h>
#include <cstdint>

#define SUB_NO 20
#define E_NO   2000
#define I_NO   500
#define T_NO   201
#define BATCH  4
#define T_DATA 20000
#define MROWS  (BATCH * T_DATA)   // 80000 rows of the projection GEMM
#define NPAD   32                 // N padded 20 -> 32 (two 16-wide WMMA tiles)
#define KC     32                 // K chunk staged in LDS per iteration
#define ASTRIDE 36                // LDS A stride (floats): 144B -> 16B-aligned rows
#define BSTRIDE 34                // LDS B stride (floats): 136B -> 8B-aligned pairs
#define VPT    4                  // async b128 loads per thread per full chunk
#define CTT    256                // t-tile for the FIR stage

typedef float v2f __attribute__((ext_vector_type(2)));
typedef float v8f __attribute__((ext_vector_type(8)));

// ---- CDNA5 async global->LDS plumbing (probe-guarded) ----------------------
#if defined(__has_builtin)
#if __has_builtin(__builtin_amdgcn_global_load_async_to_lds_b128)
#define HAVE_ASYNC_LDS 1
#endif
#endif
#ifndef HAVE_ASYNC_LDS
#define HAVE_ASYNC_LDS 0
#endif

#if defined(__has_builtin) && __has_builtin(__builtin_amdgcn_s_wait_asynccnt)
#define WAIT_ASYNC(n) __builtin_amdgcn_s_wait_asynccnt(n)
#else
#define WAIT_ASYNC(n) asm volatile("s_wait_asynccnt %0" ::"i"(n) : "memory")
#endif

#if HAVE_ASYNC_LDS
// Param 0 (per hipcc diagnostic): int __vector_size__(16) in addrspace(1)
// ("__device__").  Param 1: the LDS-side counterpart in addrspace(3).
typedef int v4i_vs __attribute__((vector_size(16)));
typedef __attribute__((address_space(1))) v4i_vs* g_v4i_p;
typedef __attribute__((address_space(3))) v4i_vs* l_v4i_p;

__device__ __forceinline__ void async_copy16(const float* g, float* l) {
  __builtin_amdgcn_global_load_async_to_lds_b128(
      (g_v4i_p)(uintptr_t)g,
      (l_v4i_p)(uint32_t)(uintptr_t)l,   // generic LDS addr: low 32 bits = LDS offset
      0, 0);
}
#endif

// ---------------------------------------------------------------------------
// Stage 1: causal FIR taps.  flip + VALID cross-corr with left pad ==
// out[t] = sum_d x[t-d] * k_unflipped[d].
// ---------------------------------------------------------------------------
__global__ void build_kern(const float* __restrict__ W_syn,
                           const float* __restrict__ Tau_syn,
                           const float* __restrict__ Delta_syn,
                           float* __restrict__ kernE,
                           float* __restrict__ kernI) {
  const int n = SUB_NO * T_NO;
  for (int idx = threadIdx.x; idx < 2 * n; idx += blockDim.x) {
    const int half = idx / n;            // 0 = excitatory, 1 = inhibitory
    const int r    = idx - half * n;
    const int s    = r / T_NO;
    const int d    = r - s * T_NO;
    const float W   = W_syn[s * 2 + half];
    const float Tau = Tau_syn[s * 2 + half];
    const float Del = Delta_syn[s * 2 + half];
    const float t_d   = fmaxf((float)d - expf(Del), 0.0f);
    const float t_tau = t_d / expf(Tau);
    const float k     = t_tau * expf(-t_tau) * expf(W);
    if (half == 0) kernE[s * T_NO + d] = k;
    else           kernI[s * T_NO + d] = k;
  }
}

// ---------------------------------------------------------------------------
// Stage 2 helpers
// ---------------------------------------------------------------------------
#if HAVE_ASYNC_LDS
// Full KC chunk: 128 x 32 floats = 1024 b128 packets, exactly VPT=4 per thread.
__device__ __forceinline__ void stage_a_async(const float* __restrict__ A, int K,
                                              int m0, int k0, float* buf, int tid) {
#pragma unroll
  for (int v = 0; v < VPT; ++v) {
    const int idx = tid + v * 256;
    const int row = idx >> 3;              // KC/4 = 8 packets per row
    const int q   = idx & 7;
    async_copy16(A + (size_t)(m0 + row) * K + k0 + (q << 2),
                 buf + row * ASTRIDE + (q << 2));
  }
}
#endif

__device__ __forceinline__ void stage_a_sync(const float* __restrict__ A, int K,
                                             int m0, int k0, int kc, float* buf,
                                             int tid) {
  const int kq = kc >> 2;
  for (int idx = tid; idx < 128 * kq; idx += 256) {
    const int row = idx / kq;
    const int q   = idx - row * kq;
    const float4 v = *reinterpret_cast<const float4*>(
        A + (size_t)(m0 + row) * K + k0 + (q << 2));
    float* p = buf + row * ASTRIDE + (q << 2);
    p[0] = v.x; p[1] = v.y; p[2] = v.z; p[3] = v.w;
  }
}

__device__ __forceinline__ void stage_b(const float* __restrict__ Csyn, int K,
                                        int k0, int kc, float* ldsB, int tid) {
  for (int idx = tid; idx < 32 * 32; idx += 256) {
    const int nn = idx >> 5;
    const int kk = idx & 31;
    float val = 0.0f;
    if (nn < SUB_NO && kk < kc) val = Csyn[(size_t)nn * K + k0 + kk];
    ldsB[nn * BSTRIDE + kk] = val;
  }
}

__device__ __forceinline__ void wmma_chunk(const float* bufA, const float* ldsB,
                                           int kc, int wave, int l16, int khalf,
                                           v8f& c0, v8f& c1) {
  const float* pa  = bufA + (wave * 16 + l16) * ASTRIDE + khalf;
  const float* pb0 = ldsB + l16 * BSTRIDE + khalf;
  const float* pb1 = ldsB + (l16 + 16) * BSTRIDE + khalf;
  for (int kk = 0; kk < kc; kk += 4) {
    const v2f a  = *reinterpret_cast<const v2f*>(pa  + kk);
    const v2f b0 = *reinterpret_cast<const v2f*>(pb0 + kk);
    const v2f b1 = *reinterpret_cast<const v2f*>(pb1 + kk);
    c0 = __builtin_amdgcn_wmma_f32_16x16x4_f32(false, a, false, b0,
                                               (short)0, c0, false, false);
    c1 = __builtin_amdgcn_wmma_f32_16x16x4_f32(false, a, false, b1,
                                               (short)0, c1, false, false);
  }
}

// ---------------------------------------------------------------------------
// Stage 2: projection GEMM  out[M, 32] = A[M, K] x Csyn^T[K, 32] via
// V_WMMA_F32_16X16X4_F32.  Full KC=32 chunks are double-buffered through LDS
// with GLOBAL_LOAD_ASYNC_TO_LDS_B128 (ASYNCcnt), tail chunk is synchronous.
// ---------------------------------------------------------------------------
__global__ __launch_bounds__(256) void proj_wmma(const float* __restrict__ A,
                                                 const float* __restrict__ Csyn,
                                                 int K,
                                                 float* __restrict__ out) {
  __shared__ __align__(16) float ldsA[2][128 * ASTRIDE];
  __shared__ __align__(16) float ldsB[32 * BSTRIDE];

  const int tid   = threadIdx.x;
  const int lane  = tid & 31;
  const int wave  = tid >> 5;
  const int l16   = lane & 15;
  const int khalf = (lane >> 4) << 1;        // lanes 16-31 hold K+2, K+3
  const int m0    = blockIdx.x * 128;

  v8f c0 = {};   // N columns 0..15
  v8f c1 = {};   // N columns 16..31

  const int nfull = K / KC;                  // E: 62, I: 15
  const int tail  = K - nfull * KC;          // E: 16, I: 20 (multiples of 4)

#if HAVE_ASYNC_LDS
  if (nfull > 0) stage_a_async(A, K, m0, 0, &ldsA[0][0], tid);
  for (int i = 0; i < nfull; ++i) {
    const int k0 = i * KC;
    float* bufA = &ldsA[i & 1][0];
    const bool more = (i + 1) < nfull;
    if (more) stage_a_async(A, K, m0, k0 + KC, &ldsA[(i + 1) & 1][0], tid);
    stage_b(Csyn, K, k0, KC, ldsB, tid);
    if (more) WAIT_ASYNC(VPT);               // chunk i retired (in-order), i+1 in flight
    else      WAIT_ASYNC(0);
    __syncthreads();
    wmma_chunk(bufA, ldsB, KC, wave, l16, khalf, c0, c1);
    __syncthreads();                         // protects buf reuse + ldsB rewrite
  }
#else
  for (int i = 0; i < nfull; ++i) {
    const int k0 = i * KC;
    float* bufA = &ldsA[i & 1][0];
    stage_a_sync(A, K, m0, k0, KC, bufA, tid);
    stage_b(Csyn, K, k0, KC, ldsB, tid);
    __syncthreads();
    wmma_chunk(bufA, ldsB, KC, wave, l16, khalf, c0, c1);
    __syncthreads();
  }
#endif

  if (tail > 0) {
    float* bufA = &ldsA[nfull & 1][0];
    stage_a_sync(A, K, m0, nfull * KC, tail, bufA, tid);
    stage_b(Csyn, K, nfull * KC, tail, ldsB, tid);
    __syncthreads();
    wmma_chunk(bufA, ldsB, tail, wave, l16, khalf, c0, c1);
  }

  // Store D: VGPR r -> M = r (lanes 0-15) / r+8 (lanes 16-31), N = lane&15.
  const int rowbase = m0 + wave * 16 + ((lane >> 4) << 3);
#pragma unroll
  for (int r = 0; r < 8; ++r) {
    out[(size_t)(rowbase + r) * NPAD + l16]      = c0[r];
    out[(size_t)(rowbase + r) * NPAD + l16 + 16] = c1[r];
  }
}

// ---------------------------------------------------------------------------
// Stage 3: 201-tap causal FIR per (b, s), e + i fused; LDS-tiled with halo.
// Intermediates (~20 MB) are L2-resident.  Writes syn transposed [s][b*T+t].
// ---------------------------------------------------------------------------
__global__ __launch_bounds__(256) void conv_kernel(const float* __restrict__ projE,
                                                   const float* __restrict__ projI,
                                                   const float* __restrict__ kernE,
                                                   const float* __restrict__ kernI,
                                                   float* __restrict__ synT) {
  __shared__ float xe[CTT + T_NO - 1];
  __shared__ float xi[CTT + T_NO - 1];
  __shared__ float ke[T_NO];
  __shared__ float ki[T_NO];

  const int tid = threadIdx.x;
  const int s   = blockIdx.y;
  const int b   = blockIdx.z;
  const int t0  = blockIdx.x * CTT;
  const size_t bt0 = (size_t)b * T_DATA;

  for (int i = tid; i < CTT + T_NO - 1; i += 256) {
    const int t = t0 + i - (T_NO - 1);
    float ve = 0.0f, vi = 0.0f;
    if (t >= 0 && t < T_DATA) {
      const size_t base = (bt0 + (size_t)t) * NPAD + s;
      ve = projE[base];
      vi = projI[base];
    }
    xe[i] = ve;
    xi[i] = vi;
  }
  for (int i = tid; i < T_NO; i += 256) {
    ke[i] = kernE[s * T_NO + i];
    ki[i] = kernI[s * T_NO + i];
  }
  __syncthreads();

  const int t = t0 + tid;
  if (t < T_DATA) {
    const int base = tid + (T_NO - 1);
    float acc = 0.0f;
#pragma unroll 4
    for (int d = 0; d < T_NO; ++d)
      acc = fmaf(xe[base - d], ke[d], fmaf(xi[base - d], ki[d], acc));
    synT[(size_t)s * MROWS + bt0 + (size_t)t] = acc;
  }
}

// ---------------------------------------------------------------------------
// Stage 4: dendritic-tree recursion, leaves -> root.  sub[k] for k <= s is
// zero when s is computed (reference semantics), so only k > s contributes.
// ---------------------------------------------------------------------------
__global__ __launch_bounds__(256) void tree_kernel(const float* __restrict__ synT,
                                                   const float* __restrict__ C_den,
                                                   const float* __restrict__ W_sub,
                                                   const float* __restrict__ Theta,
                                                   const float* __restrict__ V_o,
                                                   float* __restrict__ out) {
  const int bt = blockIdx.x * blockDim.x + threadIdx.x;
  if (bt >= MROWS) return;

  float w[SUB_NO], th[SUB_NO], sub[SUB_NO];
#pragma unroll
  for (int k = 0; k < SUB_NO; ++k) {
    w[k]   = expf(W_sub[k]);     // uniform -> scalar loads
    th[k]  = Theta[k];
    sub[k] = 0.0f;
  }
  for (int s = SUB_NO - 1; s >= 0; --s) {
    float leaf = 0.0f;
    for (int k = s + 1; k < SUB_NO; ++k)
      leaf = fmaf(C_den[s * SUB_NO + k] * w[k], sub[k], leaf);
    sub[s] = tanhf(synT[(size_t)s * MROWS + bt] + leaf + th[s]);
  }
  out[bt] = fmaf(sub[0], w[0], V_o[0]);
}

// ---------------------------------------------------------------------------
extern "C" void kernel_launch(void* const* d_in, const int* in_sizes, int n_in,
                              void* d_out, int out_size, void* d_ws, size_t ws_size,
                              hipStream_t stream) {
  (void)in_sizes; (void)n_in; (void)out_size; (void)ws_size;

  const float* S_e       = (const float*)d_in[0];
  const float* S_i       = (const float*)d_in[1];
  const float* C_den     = (const float*)d_in[2];
  const float* C_syn_e   = (const float*)d_in[3];
  const float* C_syn_i   = (const float*)d_in[4];
  const float* W_syn     = (const float*)d_in[5];
  const float* Tau_syn   = (const float*)d_in[6];
  const float* Delta_syn = (const float*)d_in[7];
  const float* W_sub     = (const float*)d_in[8];
  const float* V_o       = (const float*)d_in[9];
  const float* Theta     = (const float*)d_in[10];
  float* out = (float*)d_out;

  // Workspace layout (floats): ~26.9 MB total.
  float* ws    = (float*)d_ws;
  float* projE = ws;                                   // MROWS * 32
  float* projI = projE + (size_t)MROWS * NPAD;         // MROWS * 32
  float* synT  = projI + (size_t)MROWS * NPAD;         // SUB_NO * MROWS
  float* kernE = synT  + (size_t)SUB_NO * MROWS;       // SUB_NO * T_NO
  float* kernI = kernE + SUB_NO * T_NO;                // SUB_NO * T_NO

  build_kern<<<1, 256, 0, stream>>>(W_syn, Tau_syn, Delta_syn, kernE, kernI);

  proj_wmma<<<MROWS / 128, 256, 0, stream>>>(S_e, C_syn_e, E_NO, projE);
  proj_wmma<<<MROWS / 128, 256, 0, stream>>>(S_i, C_syn_i, I_NO, projI);

  dim3 cgrid((T_DATA + CTT - 1) / CTT, SUB_NO, BATCH);
  conv_kernel<<<cgrid, 256, 0, stream>>>(projE, projI, kernE, kernI, synT);

  tree_kernel<<<(MROWS + 255) / 256, 256, 0, stream>>>(synT, C_den, W_sub, Theta,
                                                       V_o, out);
}